// GAT_68341519613983
// MI455X (gfx1250) — compile-verified
//
#include <hip/hip_runtime.h>

// GAT 2-layer forward for MI455X (gfx1250), fp32 end-to-end.
// Matrix work: V_WMMA_F32_16X16X4_F32 fed from LDS, A-tiles staged per
// workgroup by the Tensor Data Mover (tensor_load_to_lds + s_wait_tensorcnt).

#define NEG_SLOPE 0.2f
#define BN_EPS 1e-5f

typedef __attribute__((ext_vector_type(2))) float    v2f;
typedef __attribute__((ext_vector_type(8))) float    v8f;
typedef __attribute__((ext_vector_type(4))) unsigned v4u;
typedef __attribute__((ext_vector_type(8))) int      v8i_t;
typedef __attribute__((ext_vector_type(4))) int      v4i_t;

#define LDS_STRIDE 130  // 128 dwords/row + 2-dword TDM pad (bank-conflict-free frags)

// ---------------------------------------------------------------------------
// GEMM: C[M x 128] = A[M x 128] * B[128 x 128] (fp32 WMMA 16x16x4).
// grid.x = M/16 (M multiple of 16; N=100000=6250*16), block = 256 (8 waves).
// Wave w computes the 16x16 tile for columns [16w,16w+16).
// A stripe (16x128) is DMA'd into LDS once per block by TDM with row padding
// (pad every 128 dwords by 2 dwords -> LDS row stride 130).
// A frag (16x4): lanes 0-15 K={k,k+1}, lanes 16-31 K={k+2,k+3} (ISA 7.12.2).
// B frag (4x16): mirrored; C/D: VGPR r -> row r + 8*(lane>=16), col lane&15.
// ---------------------------------------------------------------------------
__global__ __launch_bounds__(256) void gemm128_wmma_f32(
    const float* __restrict__ A, const float* __restrict__ B,
    float* __restrict__ C, int M) {
  __shared__ float sA[16 * LDS_STRIDE];
  const int lane = threadIdx.x & 31;
  const int wave = threadIdx.x >> 5;
  const int row0 = blockIdx.x << 4;
  const int col0 = wave << 4;
  const int half = lane >> 4;
  const int l15  = lane & 15;

  // Generic address of an LDS object is {SHARED_BASE, lds_offset[31:0]}
  // (ISA 10.2), so the low 32 bits are the byte offset TDM's D# wants.
  // This ptrtoint also CAPTURES sA, which the opaque asm below relies on.
  const unsigned lds_base = (unsigned)(unsigned long long)(uintptr_t)(&sA[0]);

  if (wave == 0) {
    // --- Tensor DMA descriptor (ISA 8.3/8.4): 2D tile 128 x 16, f32 ---
    const unsigned long long ga =
        (unsigned long long)(uintptr_t)(A + (size_t)row0 * 128);
    v4u g0 = { 1u,                                   // count=1 (valid user D#)
               lds_base,                             // lds_addr (bytes)
               (unsigned)(ga & 0xffffffffu),         // global_addr[31:0]
               (unsigned)((ga >> 32) & 0x01ffffffu)  // global_addr[56:32]
                   | (2u << 30) };                   // type=2 ("image")
    v8i_t g1 = {
      (int)((2u << 16)      // data_size = 4 bytes
          | (1u << 20)      // pad_enable
          | (6u << 22)      // pad_interval: code 6 -> every 128 dwords
          | (1u << 25)),    // pad_amount:   code 1 -> 2 dwords
      (int)(128u << 16),    // tensor_dim0[15:0] = 128 (bits 63:48)
      (int)0xFFFF0000u,     // tensor_dim0 hi = 0; tensor_dim1 lo16 = 0xFFFF
      (int)((128u << 16) | 0x7FFFu), // tensor_dim1 hi16 (huge), tile_dim0 = 128
      16,                   // tile_dim1 = 16, tile_dim2 = 0
      128,                  // tensor_dim0_stride = 128 elements
      0, 0 };
    v4i_t gz = { 0, 0, 0, 0 };
#if defined(__clang_major__) && (__clang_major__ >= 23)
    v8i_t gz8 = { 0, 0, 0, 0, 0, 0, 0, 0 };
    __builtin_amdgcn_tensor_load_to_lds(g0, g1, gz, gz, gz8, 0);
#else
    __builtin_amdgcn_tensor_load_to_lds(g0, g1, gz, gz, 0);
#endif
    __builtin_amdgcn_s_wait_tensorcnt(0);
  }
  __syncthreads();

  // sA is captured (ptrtoint above feeds the D#), and this opaque asm clobbers
  // memory: the compiler must assume the DMA wrote sA, so the loads below are
  // materialized -- while staying addressed off sA itself, keeping them in the
  // LDS address space (ds_load_b64, not flat_load_b64 as in round 3).
  asm volatile("" : : "s"(lds_base) : "memory");

  const float* Bcol = B + col0 + l15;
  const int abase = l15 * LDS_STRIDE;
  v8f acc = {};
#pragma unroll 4
  for (int k = 0; k < 128; k += 4) {
    const int ka = k + (half << 1);
    v2f a, b;
    a.x = sA[abase + ka];                // ds_load_b64 per k-step
    a.y = sA[abase + ka + 1];
    b.x = Bcol[(size_t)ka * 128];
    b.y = Bcol[(size_t)(ka + 1) * 128];
    acc = __builtin_amdgcn_wmma_f32_16x16x4_f32(false, a, false, b,
                                                (short)0, acc, false, false);
  }
  float* Crow = C + (size_t)(row0 + (half << 3)) * 128 + col0 + l15;
#pragma unroll
  for (int r = 0; r < 8; ++r) Crow[(size_t)r * 128] = acc[r];
}

// ---------------------------------------------------------------------------
// Per-node attention coefficients: a_s[n,h] = sum_c h[n,h,c]*att_src[h,c].
// One wave per node; lane covers 4 channels (float4); lanes 0-15 = head 0,
// lanes 16-31 = head 1; tree-reduce within 16-lane groups.
// ---------------------------------------------------------------------------
__global__ __launch_bounds__(256) void attn_coef(
    const float* __restrict__ h, const float* __restrict__ att_src,
    const float* __restrict__ att_dst, float* __restrict__ a_s,
    float* __restrict__ a_d, int n) {
  const int wid  = (int)((blockIdx.x * blockDim.x + threadIdx.x) >> 5);
  const int lane = threadIdx.x & 31;
  if (wid >= n) return;
  const float4 hv  = reinterpret_cast<const float4*>(h + (size_t)wid * 128)[lane];
  const float4 asv = reinterpret_cast<const float4*>(att_src)[lane];
  const float4 adv = reinterpret_cast<const float4*>(att_dst)[lane];
  float s = hv.x * asv.x + hv.y * asv.y + hv.z * asv.z + hv.w * asv.w;
  float d = hv.x * adv.x + hv.y * adv.y + hv.z * adv.z + hv.w * adv.w;
#pragma unroll
  for (int off = 8; off >= 1; off >>= 1) {
    s += __shfl_down(s, off, 16);
    d += __shfl_down(d, off, 16);
  }
  if ((lane & 15) == 0) {
    const int head = lane >> 4;
    a_s[wid * 2 + head] = s;
    a_d[wid * 2 + head] = d;
  }
}

// Monotonic unsigned encoding for float atomic-max.
__device__ __forceinline__ unsigned f2ord(float f) {
  unsigned u = __float_as_uint(f);
  return (u & 0x80000000u) ? ~u : (u | 0x80000000u);
}
__device__ __forceinline__ float ord2f(unsigned u) {
  return __uint_as_float((u & 0x80000000u) ? (u & 0x7fffffffu) : ~u);
}

__device__ __forceinline__ void edge_ends(int i, int nE,
                                          const int* __restrict__ esrc,
                                          const int* __restrict__ edst,
                                          int& s, int& d) {
  if (i < nE) { s = esrc[i]; d = edst[i]; }
  else        { s = i - nE;  d = i - nE; }   // appended self-loops
}

// Pass 1: segment max of leaky_relu(a_s[src]+a_d[dst]) over dst.
__global__ __launch_bounds__(256) void edge_max(
    const int* __restrict__ esrc, const int* __restrict__ edst,
    const float* __restrict__ a_s, const float* __restrict__ a_d,
    unsigned* __restrict__ mkey, int nE, int nNodes) {
  const int i = blockIdx.x * blockDim.x + threadIdx.x;
  if (i >= nE + nNodes) return;
  int s, d; edge_ends(i, nE, esrc, edst, s, d);
#pragma unroll
  for (int hh = 0; hh < 2; ++hh) {
    float e = a_s[s * 2 + hh] + a_d[d * 2 + hh];
    e = (e > 0.f) ? e : NEG_SLOPE * e;
    atomicMax(&mkey[d * 2 + hh], f2ord(e));
  }
}

// Pass 2: ex = exp(e - max[dst]); cache ex per edge; segment-sum into denom.
__global__ __launch_bounds__(256) void edge_exp(
    const int* __restrict__ esrc, const int* __restrict__ edst,
    const float* __restrict__ a_s, const float* __restrict__ a_d,
    const unsigned* __restrict__ mkey, float* __restrict__ exbuf,
    float* __restrict__ denom, int nE, int nNodes) {
  const int i = blockIdx.x * blockDim.x + threadIdx.x;
  if (i >= nE + nNodes) return;
  int s, d; edge_ends(i, nE, esrc, edst, s, d);
#pragma unroll
  for (int hh = 0; hh < 2; ++hh) {
    float e = a_s[s * 2 + hh] + a_d[d * 2 + hh];
    e = (e > 0.f) ? e : NEG_SLOPE * e;
    const float v = expf(e - ord2f(mkey[d * 2 + hh]));
    exbuf[(size_t)i * 2 + hh] = v;
    atomicAdd(&denom[d * 2 + hh], v);
  }
}

// Pass 3: out[dst] += (ex/denom[dst]) * h[src].  One wave per edge; lane owns
// 4 channels (float4 gather, 4x global_atomic_add_f32). lanes>=16 -> head 1.
__global__ __launch_bounds__(256) void edge_aggregate(
    const int* __restrict__ esrc, const int* __restrict__ edst,
    const float* __restrict__ exbuf, const float* __restrict__ denom,
    const float* __restrict__ hfeat, float* __restrict__ out,
    int nE, int nNodes) {
  const int wid  = (int)((blockIdx.x * blockDim.x + threadIdx.x) >> 5);
  const int lane = threadIdx.x & 31;
  if (wid >= nE + nNodes) return;
  int s, d; edge_ends(wid, nE, esrc, edst, s, d);
  const int head  = lane >> 4;
  const float alpha = exbuf[(size_t)wid * 2 + head] / denom[d * 2 + head];
  const float4 hv = reinterpret_cast<const float4*>(hfeat + (size_t)s * 128)[lane];
  float* o = out + (size_t)d * 128 + lane * 4;
  atomicAdd(o + 0, alpha * hv.x);
  atomicAdd(o + 1, alpha * hv.y);
  atomicAdd(o + 2, alpha * hv.z);
  atomicAdd(o + 3, alpha * hv.w);
}

// Fill n dwords with a constant bit pattern (also used for unsigned buffers).
__global__ __launch_bounds__(256) void fill_f32(float* __restrict__ p, float v, int n) {
  const int i = blockIdx.x * blockDim.x + threadIdx.x;
  if (i < n) p[i] = v;
}

// out[n*128+c] = bias[c] (pre-accumulation init with layer-1 bias).
__global__ __launch_bounds__(256) void bias_init(
    float* __restrict__ out, const float* __restrict__ b, int total) {
  const int i = blockIdx.x * blockDim.x + threadIdx.x;
  if (i < total) out[i] = b[i & 127];
}

// Per-channel sum / sum-of-squares (128 channels). blockDim=128; each thread
// owns one channel, rows strided by gridDim; two atomics per thread at the end.
__global__ __launch_bounds__(128) void bn_stats(
    const float* __restrict__ x, float* __restrict__ sums, int n) {
  const int c = threadIdx.x;
  float s = 0.f, s2 = 0.f;
  for (int r = blockIdx.x; r < n; r += gridDim.x) {
    const float v = x[(size_t)r * 128 + c];
    s += v; s2 += v * v;
  }
  atomicAdd(&sums[c], s);
  atomicAdd(&sums[128 + c], s2);
}

// In-place BatchNorm (biased var) + ELU.
__global__ __launch_bounds__(256) void bn_elu(
    float* __restrict__ x, const float* __restrict__ sums,
    const float* __restrict__ gamma, const float* __restrict__ beta, int n) {
  const int i = blockIdx.x * blockDim.x + threadIdx.x;
  if (i >= n * 128) return;
  const int c = i & 127;
  const float inv_n = 1.0f / (float)n;
  const float mu  = sums[c] * inv_n;
  const float var = sums[128 + c] * inv_n - mu * mu;
  const float v = gamma[c] * (x[i] - mu) * rsqrtf(var + BN_EPS) + beta[c];
  x[i] = (v > 0.f) ? v : (expf(v) - 1.f);
}

// out[n,c] = mean over 2 heads of agg[n,h,c] + b2[c].
__global__ __launch_bounds__(256) void head_mean_bias(
    const float* __restrict__ agg, const float* __restrict__ b2,
    float* __restrict__ out, int n) {
  const int i = blockIdx.x * blockDim.x + threadIdx.x;
  if (i >= n * 64) return;
  const int node = i >> 6;
  const int c = i & 63;
  out[i] = 0.5f * (agg[(size_t)node * 128 + c] + agg[(size_t)node * 128 + 64 + c]) + b2[c];
}

extern "C" void kernel_launch(void* const* d_in, const int* in_sizes, int n_in,
                              void* d_out, int out_size, void* d_ws, size_t ws_size,
                              hipStream_t stream) {
  const float* x        = (const float*)d_in[0];
  const int*   ei       = (const int*)  d_in[1];
  const float* W1       = (const float*)d_in[2];
  const float* att_src1 = (const float*)d_in[3];
  const float* att_dst1 = (const float*)d_in[4];
  const float* b1       = (const float*)d_in[5];
  const float* gamma1   = (const float*)d_in[6];
  const float* beta1    = (const float*)d_in[7];
  const float* W2       = (const float*)d_in[8];
  const float* att_src2 = (const float*)d_in[9];
  const float* att_dst2 = (const float*)d_in[10];
  const float* b2       = (const float*)d_in[11];
  float* out = (float*)d_out;

  const int nNodes = in_sizes[0] / 128;     // 100000
  const int nE     = in_sizes[1] / 2;       // 1600000
  const int total  = nE + nNodes;           // edges + self-loops
  const int* esrc = ei;
  const int* edst = ei + nE;

  // Workspace carve-up (all fp32 words): ~120 MB.
  float*    h     = (float*)d_ws;                          // [N,128] h1 then h2
  float*    agg   = h + (size_t)nNodes * 128;              // [N,128] out1/act/out2
  float*    a_s   = agg + (size_t)nNodes * 128;            // [N,2]
  float*    a_d   = a_s + (size_t)nNodes * 2;              // [N,2]
  unsigned* mkey  = (unsigned*)(a_d + (size_t)nNodes * 2); // [N,2]
  float*    denom = (float*)(mkey + (size_t)nNodes * 2);   // [N,2]
  float*    exbuf = denom + (size_t)nNodes * 2;            // [total,2]
  float*    sums  = exbuf + (size_t)total * 2;             // [256]

  const int B = 256;
  const int gemmBlocks = (nNodes + 15) / 16;               // M multiple of 16
  const int nodeWaveBlocks = (nNodes * 32 + B - 1) / B;
  const int edgeBlocks = (total + B - 1) / B;
  const int edgeWaveBlocks = (total + 7) / 8;              // total*32/256
  const int n2Blocks  = (nNodes * 2 + B - 1) / B;
  const int nfBlocks  = (nNodes * 128 + B - 1) / B;
  const int noBlocks  = (nNodes * 64 + B - 1) / B;

  // ---------------- Layer 1 (concat heads) ----------------
  gemm128_wmma_f32<<<gemmBlocks, B, 0, stream>>>(x, W1, h, nNodes);
  attn_coef<<<nodeWaveBlocks, B, 0, stream>>>(h, att_src1, att_dst1, a_s, a_d, nNodes);
  fill_f32<<<n2Blocks, B, 0, stream>>>((float*)mkey, 0.0f, nNodes * 2); // 0u < all keys
  fill_f32<<<n2Blocks, B, 0, stream>>>(denom, 0.0f, nNodes * 2);
  edge_max<<<edgeBlocks, B, 0, stream>>>(esrc, edst, a_s, a_d, mkey, nE, nNodes);
  edge_exp<<<edgeBlocks, B, 0, stream>>>(esrc, edst, a_s, a_d, mkey, exbuf, denom, nE, nNodes);
  bias_init<<<nfBlocks, B, 0, stream>>>(agg, b1, nNodes * 128);
  edge_aggregate<<<edgeWaveBlocks, B, 0, stream>>>(esrc, edst, exbuf, denom, h, agg, nE, nNodes);

  // BatchNorm + ELU (in place on agg).
  fill_f32<<<1, B, 0, stream>>>(sums, 0.0f, 256);
  bn_stats<<<512, 128, 0, stream>>>(agg, sums, nNodes);
  bn_elu<<<nfBlocks, B, 0, stream>>>(agg, sums, gamma1, beta1, nNodes);

  // ---------------- Layer 2 (mean over heads) ----------------
  gemm128_wmma_f32<<<gemmBlocks, B, 0, stream>>>(agg, W2, h, nNodes);
  attn_coef<<<nodeWaveBlocks, B, 0, stream>>>(h, att_src2, att_dst2, a_s, a_d, nNodes);
  fill_f32<<<n2Blocks, B, 0, stream>>>((float*)mkey, 0.0f, nNodes * 2);
  fill_f32<<<n2Blocks, B, 0, stream>>>(denom, 0.0f, nNodes * 2);
  edge_max<<<edgeBlocks, B, 0, stream>>>(esrc, edst, a_s, a_d, mkey, nE, nNodes);
  edge_exp<<<edgeBlocks, B, 0, stream>>>(esrc, edst, a_s, a_d, mkey, exbuf, denom, nE, nNodes);
  fill_f32<<<nfBlocks, B, 0, stream>>>(agg, 0.0f, nNodes * 128);  // agg now out2
  edge_aggregate<<<edgeWaveBlocks, B, 0, stream>>>(esrc, edst, exbuf, denom, h, agg, nE, nNodes);
  head_mean_bias<<<noBlocks, B, 0, stream>>>(agg, b2, out, nNodes);
}